// Grid_5875515261577
// MI455X (gfx1250) — compile-verified
//
#include <hip/hip_runtime.h>

// ---------------------------------------------------------------------------
// Cloud-in-cell (CIC) deposition of 10M weighted particles onto a 256^3 grid.
//
// MI455X reasoning (compile-only, so reasoned from specs):
//  * 80M fp32 scatter atomics dominate; the 67MB grid fits in the 192MB L2,
//    so global_atomic_add_f32 (no-return, STOREcnt-tracked) is absorbed at L2.
//  * The 160MB particle stream is read once -> non-temporal loads so it does
//    not evict the grid working set from L2.
//  * Grid index packed as (ix<<16)|(iy<<8)|iz (32-bit); all 8 corner offsets
//    fit the 24-bit instruction IOFFSET, so the 8 atomics share one VGPR
//    offset against the SGPR base (GVS addressing) -> no 64-bit address math.
//  * Deposit with per-corner bounds checks instead of the reference's
//    pad-by-1-and-crop (identical result, no 69MB scratch grid).
// ---------------------------------------------------------------------------

namespace {
constexpr int   kN       = 256;                       // N_CELLS
constexpr float kGridMin = -10.0f;
constexpr float kDx      = 20.0f / 255.0f;            // (max-min)/(n-1), fp32
}

__device__ __forceinline__ void atomic_add_f32(float* p, float v) {
  // Lowers to global_atomic_add_f32 (no return value) on CDNA5.
  unsafeAtomicAdd(p, v);
}

__global__ __launch_bounds__(256) void cic_deposit_kernel(
    const float* __restrict__ pos,   // [n,3] row-major
    const float* __restrict__ wgt,   // [n]
    float* __restrict__ grid,        // [256,256,256] flat
    int n) {
  int i = blockIdx.x * blockDim.x + threadIdx.x;
  if (i >= n) return;

  // Streaming, read-once data: non-temporal so the grid stays hot in L2.
  size_t pbase = 3u * (size_t)i;
  float px = __builtin_nontemporal_load(pos + pbase + 0);
  float py = __builtin_nontemporal_load(pos + pbase + 1);
  float pz = __builtin_nontemporal_load(pos + pbase + 2);
  float w  = __builtin_nontemporal_load(wgt + i);

  // Float cell index, matching reference arithmetic (true fp32 divide by DX).
  float fx = (px - kGridMin) / kDx;
  float fy = (py - kGridMin) / kDx;
  float fz = (pz - kGridMin) / kDx;

  float gx = floorf(fx), gy = floorf(fy), gz = floorf(fz);
  int   ix = (int)gx,    iy = (int)gy,    iz = (int)gz;
  float ox = fx - gx,    oy = fy - gy,    oz = fz - gz;

  // Reference in-grid mask: padded idx=floor+1 in [1,256]  <=>  floor in [0,255].
  if ((unsigned)ix > 255u || (unsigned)iy > 255u || (unsigned)iz > 255u) return;

  float wx0 = 1.0f - ox, wx1 = ox;
  float wy0 = 1.0f - oy, wy1 = oy;
  float wz0 = 1.0f - oz, wz1 = oz;

  // +1 corners at unpadded index 256 correspond to padded index 257, which the
  // reference crops away -> drop them.
  bool x1 = (ix + 1) < kN;
  bool y1 = (iy + 1) < kN;
  bool z1 = (iz + 1) < kN;

  // 32-bit flat index; element offsets from base: z:+1, y:+256, x:+65536.
  unsigned base = ((unsigned)ix << 16) | ((unsigned)iy << 8) | (unsigned)iz;
  float* g = grid + base;

  float w00 = w * wx0 * wy0;
  atomic_add_f32(g, w00 * wz0);
  if (z1) atomic_add_f32(g + 1, w00 * wz1);

  if (y1) {
    float w01 = w * wx0 * wy1;
    atomic_add_f32(g + 256, w01 * wz0);
    if (z1) atomic_add_f32(g + 257, w01 * wz1);
  }

  if (x1) {
    float w10 = w * wx1 * wy0;
    atomic_add_f32(g + 65536, w10 * wz0);
    if (z1) atomic_add_f32(g + 65537, w10 * wz1);

    if (y1) {
      float w11 = w * wx1 * wy1;
      atomic_add_f32(g + 65536 + 256, w11 * wz0);
      if (z1) atomic_add_f32(g + 65536 + 257, w11 * wz1);
    }
  }
}

// Zero the output grid (harness poisons d_out; must re-zero every call).
__global__ __launch_bounds__(256) void zero_grid_kernel(float4* __restrict__ p,
                                                        int n4) {
  int i = blockIdx.x * blockDim.x + threadIdx.x;
  int stride = gridDim.x * blockDim.x;
  for (; i < n4; i += stride) {
    p[i] = float4{0.0f, 0.0f, 0.0f, 0.0f};
  }
}

extern "C" void kernel_launch(void* const* d_in, const int* in_sizes, int n_in,
                              void* d_out, int out_size, void* d_ws, size_t ws_size,
                              hipStream_t stream) {
  const float* pos = (const float*)d_in[0];   // [N,3] float32
  const float* wgt = (const float*)d_in[1];   // [N]   float32
  float* grid = (float*)d_out;                // 256^3 float32
  int n = in_sizes[1];                        // particle count (weights length)

  // 1) Zero the grid (out_size = 256^3 = 16,777,216, divisible by 4).
  int n4 = out_size / 4;
  zero_grid_kernel<<<4096, 256, 0, stream>>>((float4*)grid, n4);

  // 2) Deposit.
  int threads = 256;
  int blocks = (n + threads - 1) / threads;
  cic_deposit_kernel<<<blocks, threads, 0, stream>>>(pos, wgt, grid, n);
}